// AttentionLayer_55207509622987
// MI455X (gfx1250) — compile-verified
//
#include <hip/hip_runtime.h>
#include <math.h>

// Problem constants (match the reference)
#define S_LEN 4096
#define BATCH 64
#define HID   512
#define NC    16            // number of S-chunks in the streaming pass
#define CHUNK (S_LEN / NC)  // 256 s-values per workgroup

typedef __attribute__((ext_vector_type(2))) float v2f;
typedef __attribute__((ext_vector_type(8))) float v8f;

// ---------------------------------------------------------------------------
// Kernel 1: q = h_t @ W_a   ([64x512] x [512x512] -> [64x512]) via f32 WMMA.
// Tiles: M=4 x N=32 = 128 tiles of 16x16, one wave each. Block = 4 waves.
// A fragment (16x4 f32, 2 VGPR): lanes 0-15 hold K=0,1; lanes 16-31 hold K=2,3.
// B fragment (4x16 f32, 2 VGPR): lanes hold column N=lane%16, same K split.
// D (16x16 f32, 8 VGPR): vgpr r, lanes 0-15 -> M=r; lanes 16-31 -> M=8+r.
// ---------------------------------------------------------------------------
__global__ void qproj_wmma(const float* __restrict__ h_t,
                           const float* __restrict__ W_a,
                           float* __restrict__ q) {
  const int wave = threadIdx.x >> 5;
  const int tile = blockIdx.x * 4 + wave;       // 0..127
  const int tn = tile & 31;                     // N tile (0..31)
  const int tm = tile >> 5;                     // M tile (0..3)
  const int l   = threadIdx.x & 31;
  const int row = l & 15;                       // A row / B col within tile
  const int kh  = l >> 4;                       // K-half select

  const float* Arow = h_t + (size_t)(tm * 16 + row) * HID;
  const int    ncol = tn * 16 + row;

  v8f acc = {};
  for (int k = 0; k < HID; k += 4) {
    const int k2 = k + 2 * kh;
    const float2 av = *(const float2*)(Arow + k2);
    v2f a; a.x = av.x; a.y = av.y;
    v2f b;
    b.x = W_a[(size_t)k2 * HID + ncol];
    b.y = W_a[(size_t)(k2 + 1) * HID + ncol];
    acc = __builtin_amdgcn_wmma_f32_16x16x4_f32(false, a, false, b,
                                                (short)0, acc, false, false);
  }
  #pragma unroll
  for (int r = 0; r < 8; ++r) {
    const int mr = tm * 16 + kh * 8 + r;
    q[(size_t)mr * HID + ncol] = acc[r];
  }
}

// ---------------------------------------------------------------------------
// Kernel 1b: c0[b] = dot(b_a, h_t[b])  (bias folded into scores)
// ---------------------------------------------------------------------------
__global__ void bias_dot(const float* __restrict__ h_t,
                         const float* __restrict__ b_a,
                         float* __restrict__ c0) {
  const int b = blockIdx.x;
  const int lane = threadIdx.x;
  float s = 0.f;
  for (int h = lane; h < HID; h += 32)
    s += h_t[(size_t)b * HID + h] * b_a[h];
  #pragma unroll
  for (int off = 16; off > 0; off >>= 1)
    s += __shfl_xor(s, off, 32);
  if (lane == 0) c0[b] = s;
}

// ---------------------------------------------------------------------------
// Kernel 2: fused streaming pass with online softmax.
// grid = (BATCH, NC), block = 256 (8 waves). Each wave owns a strided subset
// of the chunk's s-values; per s it loads hs[s,b,:] once (float4, coalesced),
// computes score = hs.q + c0 via wave-reduced dot, and accumulates
// exp-weighted context in registers with online rescaling.
// Per-workgroup partial (m, l, c[512]) written to workspace.
// ---------------------------------------------------------------------------
__global__ void flash_pass(const float* __restrict__ hs,
                           const float* __restrict__ q,
                           const float* __restrict__ c0,
                           const int* __restrict__ mask,   // [B,S]
                           float* __restrict__ part_m,
                           float* __restrict__ part_l,
                           float* __restrict__ part_c) {
  const int b = blockIdx.x;
  const int chunk = blockIdx.y;
  const int tid = threadIdx.x;
  const int w = tid >> 5;
  const int lane = tid & 31;

  // q fragment: lane holds h = lane*4 + j*128 + {0..3}
  float4 qv[4];
  #pragma unroll
  for (int j = 0; j < 4; ++j)
    qv[j] = *(const float4*)(q + (size_t)b * HID + lane * 4 + j * 128);
  const float c0b = c0[b];

  float m = -INFINITY, lsum = 0.f;
  float4 cacc[4];
  #pragma unroll
  for (int j = 0; j < 4; ++j) cacc[j] = make_float4(0.f, 0.f, 0.f, 0.f);

  const int s0 = chunk * CHUNK + w;
  for (int it = 0; it < CHUNK / 8; ++it) {
    const int s = s0 + it * 8;
    if (mask[(size_t)b * S_LEN + s] == 0) continue;   // log(0) = -inf -> weight 0
    const float* hp = hs + ((size_t)s * BATCH + b) * HID;
    float4 hv[4];
    #pragma unroll
    for (int j = 0; j < 4; ++j)
      hv[j] = *(const float4*)(hp + lane * 4 + j * 128);

    float d = 0.f;
    #pragma unroll
    for (int j = 0; j < 4; ++j)
      d += hv[j].x * qv[j].x + hv[j].y * qv[j].y +
           hv[j].z * qv[j].z + hv[j].w * qv[j].w;
    #pragma unroll
    for (int off = 16; off > 0; off >>= 1)
      d += __shfl_xor(d, off, 32);

    const float score = d + c0b;
    const float nm = fmaxf(m, score);
    const float sc = __expf(m - nm);     // exp(-inf)=0 on first hit
    const float p  = __expf(score - nm);
    lsum = lsum * sc + p;
    #pragma unroll
    for (int j = 0; j < 4; ++j) {
      cacc[j].x = cacc[j].x * sc + p * hv[j].x;
      cacc[j].y = cacc[j].y * sc + p * hv[j].y;
      cacc[j].z = cacc[j].z * sc + p * hv[j].z;
      cacc[j].w = cacc[j].w * sc + p * hv[j].w;
    }
    m = nm;
  }

  // Combine the 8 waves' partials via LDS.
  __shared__ __align__(16) float s_c[8 * HID];
  __shared__ float s_m[8], s_l[8];
  #pragma unroll
  for (int j = 0; j < 4; ++j)
    *(float4*)(s_c + w * HID + lane * 4 + j * 128) = cacc[j];
  if (lane == 0) { s_m[w] = m; s_l[w] = lsum; }
  __syncthreads();

  float M = -INFINITY;
  #pragma unroll
  for (int ww = 0; ww < 8; ++ww) M = fmaxf(M, s_m[ww]);
  float L = 0.f, a0 = 0.f, a1 = 0.f;
  #pragma unroll
  for (int ww = 0; ww < 8; ++ww) {
    const float scw = (s_m[ww] == -INFINITY) ? 0.f : __expf(s_m[ww] - M);
    L  += scw * s_l[ww];
    a0 += scw * s_c[ww * HID + tid];
    a1 += scw * s_c[ww * HID + tid + 256];
  }
  const int pi = b * NC + chunk;
  part_c[(size_t)pi * HID + tid]       = a0;
  part_c[(size_t)pi * HID + tid + 256] = a1;
  if (tid == 0) { part_m[pi] = M; part_l[pi] = L; }
}

// ---------------------------------------------------------------------------
// Kernel 3: merge chunk partials per batch -> cat = [c_t , h_t]  ([B, 2H])
// ---------------------------------------------------------------------------
__global__ void combine(const float* __restrict__ part_m,
                        const float* __restrict__ part_l,
                        const float* __restrict__ part_c,
                        const float* __restrict__ h_t,
                        float* __restrict__ cat) {
  const int b = blockIdx.x;
  const int t = threadIdx.x;   // 256 threads

  float M = -INFINITY;
  for (int i = 0; i < NC; ++i) M = fmaxf(M, part_m[b * NC + i]);
  float scale[NC];
  float L = 0.f;
  for (int i = 0; i < NC; ++i) {
    const float mi = part_m[b * NC + i];
    scale[i] = (mi == -INFINITY) ? 0.f : __expf(mi - M);
    L += scale[i] * part_l[b * NC + i];
  }
  const float inv = (L > 0.f) ? (1.f / L) : 0.f;

  for (int h = t; h < HID; h += 256) {
    float acc = 0.f;
    for (int i = 0; i < NC; ++i)
      acc += scale[i] * part_c[(size_t)(b * NC + i) * HID + h];
    cat[(size_t)b * (2 * HID) + h]       = acc * inv;
    cat[(size_t)b * (2 * HID) + HID + h] = h_t[(size_t)b * HID + h];
  }
}

// ---------------------------------------------------------------------------
// Kernel 4: out = tanh(cat @ W_r^T + b_r)   ([64x1024] x [1024x512]^T-view)
// Same WMMA tiling as kernel 1; B fragment reads W_r rows (contiguous float2).
// ---------------------------------------------------------------------------
__global__ void outproj_wmma(const float* __restrict__ cat,
                             const float* __restrict__ W_r,
                             const float* __restrict__ b_r,
                             float* __restrict__ out) {
  const int wave = threadIdx.x >> 5;
  const int tile = blockIdx.x * 4 + wave;       // 0..127
  const int tn = tile & 31;
  const int tm = tile >> 5;
  const int l   = threadIdx.x & 31;
  const int row = l & 15;
  const int kh  = l >> 4;

  const float* Arow = cat + (size_t)(tm * 16 + row) * (2 * HID);
  const float* Bcol = W_r + (size_t)(tn * 16 + row) * (2 * HID); // W_r[n, :]
  const int    ncol = tn * 16 + row;

  v8f acc = {};
  for (int k = 0; k < 2 * HID; k += 4) {
    const int k2 = k + 2 * kh;
    const float2 av = *(const float2*)(Arow + k2);
    const float2 bv = *(const float2*)(Bcol + k2);   // B[k2][n]=W_r[n][k2]
    v2f a; a.x = av.x; a.y = av.y;
    v2f b; b.x = bv.x; b.y = bv.y;
    acc = __builtin_amdgcn_wmma_f32_16x16x4_f32(false, a, false, b,
                                                (short)0, acc, false, false);
  }
  const float bias = b_r[ncol];
  #pragma unroll
  for (int r = 0; r < 8; ++r) {
    const int mr = tm * 16 + kh * 8 + r;
    out[(size_t)mr * HID + ncol] = tanhf(acc[r] + bias);
  }
}

// ---------------------------------------------------------------------------
extern "C" void kernel_launch(void* const* d_in, const int* in_sizes, int n_in,
                              void* d_out, int out_size, void* d_ws, size_t ws_size,
                              hipStream_t stream) {
  const float* hs   = (const float*)d_in[0];   // [S,B,H]
  const float* h_t  = (const float*)d_in[1];   // [B,H]
  const int*   mask = (const int*)  d_in[2];   // [B,S]
  const float* W_a  = (const float*)d_in[3];   // [H,H]
  const float* b_a  = (const float*)d_in[4];   // [H]
  const float* W_r  = (const float*)d_in[5];   // [H,2H]
  const float* b_r  = (const float*)d_in[6];   // [H]
  float* out = (float*)d_out;                  // [B,H]

  // Workspace layout (floats): ~2.5 MB total
  float* ws = (float*)d_ws;
  float* q      = ws;                          // B*H
  float* c0     = q  + BATCH * HID;            // B
  float* pm     = c0 + BATCH;                  // B*NC
  float* pl     = pm + BATCH * NC;             // B*NC
  float* pc     = pl + BATCH * NC;             // B*NC*H
  float* catbuf = pc + (size_t)BATCH * NC * HID; // B*2H

  qproj_wmma  <<<32, 128, 0, stream>>>(h_t, W_a, q);
  bias_dot    <<<BATCH, 32, 0, stream>>>(h_t, b_a, c0);
  flash_pass  <<<dim3(BATCH, NC), 256, 0, stream>>>(hs, q, c0, mask, pm, pl, pc);
  combine     <<<BATCH, 256, 0, stream>>>(pm, pl, pc, h_t, catbuf);
  outproj_wmma<<<32, 128, 0, stream>>>(catbuf, W_r, b_r, out);
}